// GATModel_670014898852
// MI455X (gfx1250) — compile-verified
//
#include <hip/hip_runtime.h>

// ---------------------------------------------------------------------------
// GAT 2-layer forward for MI455X (gfx1250, wave32).
// GEMMs use native fp32 WMMA (V_WMMA_F32_16X16X4_F32) -- exact fp32, and the
// GEMMs are a small fraction of runtime (workload is L2-atomic bound), so no
// precision downconversion is worthwhile.
// ---------------------------------------------------------------------------

typedef __attribute__((ext_vector_type(2))) float v2f;
typedef __attribute__((ext_vector_type(8))) float v8f;

#define N_NODES 50000
#define N_EDGES 800000
#define E_TOT   (N_EDGES + N_NODES)   // with self loops
#define IN_DIM  128
#define HID     256
#define HEADS   4
#define C1      64                    // HID / HEADS
#define OUT_DIM 128
#define NEG_SLOPE 0.2f

// Order-preserving float -> uint encoding for atomic max on floats.
__device__ __forceinline__ unsigned f2ord(float x) {
    unsigned u = __float_as_uint(x);
    return (u & 0x80000000u) ? ~u : (u | 0x80000000u);
}
__device__ __forceinline__ float ord2f(unsigned e) {
    return (e & 0x80000000u) ? __uint_as_float(e & 0x7fffffffu)
                             : __uint_as_float(~e);
}
#define ENC_NEGINF 0x007FFFFFu   // f2ord(-inf)

// ---------------------------------------------------------------------------
__global__ void fill_u32(unsigned* __restrict__ p, unsigned v, int n) {
    int i = blockIdx.x * blockDim.x + threadIdx.x;
    int stride = gridDim.x * blockDim.x;
    for (; i < n; i += stride) p[i] = v;
}

// ---------------------------------------------------------------------------
// C[M,N] = A[M,K] * B[K,N], fp32, wave-per-16x16 output tile via
// v_wmma_f32_16x16x4_f32.  M % 16 == 0, N % 16 == 0, K % 4 == 0.
// A-frag (16x4):  lanes 0-15 rows M0..15 with K = k..k+1, lanes 16-31 K = k+2..k+3
// B-frag (4x16):  VGPR0 = row (k + 2*(lane>>4)), VGPR1 = next row; cols = lane&15
// C/D (16x16):    vgpr r -> row r + 8*(lane>>4), col = lane&15
__global__ void gemm_wmma_f32(const float* __restrict__ A,
                              const float* __restrict__ B,
                              float* __restrict__ C,
                              int M, int K, int N) {
    int gwave = (blockIdx.x * blockDim.x + threadIdx.x) >> 5;
    int lane  = threadIdx.x & 31;
    int ntiles = N >> 4;
    int mt = gwave / ntiles;
    int nt = gwave - mt * ntiles;
    if (mt * 16 >= M) return;                       // uniform per wave

    const int half  = lane >> 4;                    // 0 or 1
    const int l15   = lane & 15;
    const int row   = mt * 16 + l15;                // A row for this lane
    const int col   = nt * 16 + l15;                // B/C col for this lane
    const int koff  = half * 2;

    v8f acc = {};
    const float* arow = A + (long)row * K;
    for (int k = 0; k < K; k += 4) {
        v2f a, b;
        a.x = arow[k + koff];
        a.y = arow[k + koff + 1];
        b.x = B[(long)(k + koff) * N + col];
        b.y = B[(long)(k + koff + 1) * N + col];
        acc = __builtin_amdgcn_wmma_f32_16x16x4_f32(
            false, a, false, b, (short)0, acc, false, false);
    }
    int r0 = mt * 16 + half * 8;
#pragma unroll
    for (int r = 0; r < 8; ++r)
        C[(long)(r0 + r) * N + col] = acc[r];
}

// ---------------------------------------------------------------------------
// Per-(node,head) attention logits: as[n,h] = dot(hfeat[n,h,:], a_src[h,:])
__global__ void alph_kernel(const float* __restrict__ h,
                            const float* __restrict__ a_src,
                            const float* __restrict__ a_dst,
                            float* __restrict__ as_o,
                            float* __restrict__ ad_o,
                            int nn, int H, int C) {
    int i = blockIdx.x * blockDim.x + threadIdx.x;
    if (i >= nn * H) return;
    int hh = i % H;
    const float* hp  = h + (long)i * C;       // (n*H + hh)*C : HID contiguous
    const float* asp = a_src + hh * C;
    const float* adp = a_dst + hh * C;
    float s = 0.f, d = 0.f;
    for (int c = 0; c < C; ++c) {
        float v = hp[c];
        s += v * asp[c];
        d += v * adp[c];
    }
    as_o[i] = s;
    ad_o[i] = d;
}

__device__ __forceinline__ void edge_ids(int e, const int* __restrict__ src,
                                         const int* __restrict__ dst,
                                         int& s, int& d) {
    if (e < N_EDGES) { s = src[e]; d = dst[e]; }
    else             { s = e - N_EDGES; d = s; }   // self loop
}

// Pass 1: segment max over dst (ordered-uint hardware atomic max)
__global__ void edge_max(const float* __restrict__ as_v,
                         const float* __restrict__ ad_v,
                         const int* __restrict__ src,
                         const int* __restrict__ dst,
                         unsigned* __restrict__ mEnc, int H) {
    int e = blockIdx.x * blockDim.x + threadIdx.x;
    if (e >= E_TOT) return;
    int s, d; edge_ids(e, src, dst, s, d);
    for (int h = 0; h < H; ++h) {
        float v = as_v[s * H + h] + ad_v[d * H + h];
        v = (v > 0.f) ? v : NEG_SLOPE * v;
        atomicMax(&mEnc[d * H + h], f2ord(v));
    }
}

// Pass 2: ex = exp(e - m[dst]); cache per edge; segment sum via hw f32 atomic
__global__ void edge_exp(const float* __restrict__ as_v,
                         const float* __restrict__ ad_v,
                         const int* __restrict__ src,
                         const int* __restrict__ dst,
                         const unsigned* __restrict__ mEnc,
                         float* __restrict__ ssum,
                         float* __restrict__ exbuf, int H) {
    int e = blockIdx.x * blockDim.x + threadIdx.x;
    if (e >= E_TOT) return;
    int s, d; edge_ids(e, src, dst, s, d);
    for (int h = 0; h < H; ++h) {
        float v = as_v[s * H + h] + ad_v[d * H + h];
        v = (v > 0.f) ? v : NEG_SLOPE * v;
        float ex = __expf(v - ord2f(mEnc[d * H + h]));
        exbuf[(long)e * H + h] = ex;
        unsafeAtomicAdd(&ssum[d * H + h], ex);     // global_atomic_add_f32
    }
}

// Pass 3: out[dst, :] += h[src, :] * alpha   (block per edge, thread per chan)
__global__ void edge_agg(const float* __restrict__ hfeat,
                         const int* __restrict__ src,
                         const int* __restrict__ dst,
                         const float* __restrict__ exbuf,
                         const float* __restrict__ ssum,
                         float* __restrict__ out, int H, int C) {
    int e = blockIdx.x;
    int c = threadIdx.x;                // 0 .. H*C-1
    int s, d; edge_ids(e, src, dst, s, d);
    int h = c / C;
    float alpha = exbuf[(long)e * H + h] / (ssum[d * H + h] + 1e-16f);
    float v = hfeat[(long)s * (H * C) + c] * alpha;
    unsafeAtomicAdd(&out[(long)d * (H * C) + c], v);
}

__global__ void bias_relu_inplace(float* __restrict__ p,
                                  const float* __restrict__ b, int n, int D) {
    int i = blockIdx.x * blockDim.x + threadIdx.x;
    if (i >= n) return;
    float v = p[i] + b[i % D];
    p[i] = v > 0.f ? v : 0.f;
}

__global__ void bias_add_inplace(float* __restrict__ p,
                                 const float* __restrict__ b, int n, int D) {
    int i = blockIdx.x * blockDim.x + threadIdx.x;
    if (i >= n) return;
    p[i] = p[i] + b[i % D];
}

// ---------------------------------------------------------------------------
extern "C" void kernel_launch(void* const* d_in, const int* in_sizes, int n_in,
                              void* d_out, int out_size, void* d_ws, size_t ws_size,
                              hipStream_t stream) {
    const float* x    = (const float*)d_in[0];
    const int*   ei   = (const int*)d_in[1];     // [2, N_EDGES] int32
    const float* W1   = (const float*)d_in[2];
    const float* a1s  = (const float*)d_in[3];
    const float* a1d  = (const float*)d_in[4];
    const float* b1   = (const float*)d_in[5];
    const float* W2   = (const float*)d_in[6];
    const float* a2s  = (const float*)d_in[7];
    const float* a2d  = (const float*)d_in[8];
    const float* b2   = (const float*)d_in[9];
    float* out = (float*)d_out;

    const int* src = ei;
    const int* dst = ei + N_EDGES;

    // ---- workspace layout (floats) ----
    float* ws = (float*)d_ws;
    const long SZ_H1  = (long)N_NODES * HID;       // 12.8M
    float*    h1   = ws;                           // layer1 gemm out; reused: g2
    float*    agg1 = ws + SZ_H1;                   // layer1 agg -> h2 in place
    float*    exb  = ws + 2 * SZ_H1;               // E_TOT*HEADS exp cache
    float*    tail = ws + 2 * SZ_H1 + (long)E_TOT * HEADS;
    float*    as1  = tail;                         // N*4
    float*    ad1  = as1 + (long)N_NODES * HEADS;
    unsigned* m1   = (unsigned*)(ad1 + (long)N_NODES * HEADS);
    float*    s1   = (float*)(m1 + (long)N_NODES * HEADS);
    float*    as2  = s1 + (long)N_NODES * HEADS;
    float*    ad2  = as2 + N_NODES;
    unsigned* m2   = (unsigned*)(ad2 + N_NODES);
    float*    s2   = (float*)(m2 + N_NODES);
    float*    g2   = h1;                           // layer2 gemm out (alias)

    const int TB = 256;
    auto cdiv = [](long a, long b) { return (int)((a + b - 1) / b); };

    // ---- init accumulators ----
    fill_u32<<<1024, TB, 0, stream>>>((unsigned*)agg1, 0u, (int)SZ_H1);
    fill_u32<<<1024, TB, 0, stream>>>((unsigned*)out, 0u, N_NODES * OUT_DIM);
    fill_u32<<<cdiv(N_NODES * HEADS, TB), TB, 0, stream>>>(m1, ENC_NEGINF, N_NODES * HEADS);
    fill_u32<<<cdiv(N_NODES * HEADS, TB), TB, 0, stream>>>((unsigned*)s1, 0u, N_NODES * HEADS);
    fill_u32<<<cdiv(N_NODES, TB), TB, 0, stream>>>(m2, ENC_NEGINF, N_NODES);
    fill_u32<<<cdiv(N_NODES, TB), TB, 0, stream>>>((unsigned*)s2, 0u, N_NODES);

    // ---- layer 1 ----
    {   // GEMM1: [50000,128] x [128,256] ; 3125*16 tiles, 8 waves/block
        long waves = (long)(N_NODES / 16) * (HID / 16);
        gemm_wmma_f32<<<cdiv(waves * 32, TB), TB, 0, stream>>>(
            x, W1, h1, N_NODES, IN_DIM, HID);
    }
    alph_kernel<<<cdiv((long)N_NODES * HEADS, TB), TB, 0, stream>>>(
        h1, a1s, a1d, as1, ad1, N_NODES, HEADS, C1);
    edge_max<<<cdiv(E_TOT, TB), TB, 0, stream>>>(as1, ad1, src, dst, m1, HEADS);
    edge_exp<<<cdiv(E_TOT, TB), TB, 0, stream>>>(as1, ad1, src, dst, m1, s1, exb, HEADS);
    edge_agg<<<E_TOT, HEADS * C1, 0, stream>>>(h1, src, dst, exb, s1, agg1, HEADS, C1);
    bias_relu_inplace<<<cdiv(SZ_H1, TB), TB, 0, stream>>>(agg1, b1, (int)SZ_H1, HID);

    // ---- layer 2 (agg1 now holds h2) ----
    {   // GEMM2: [50000,256] x [256,128]
        long waves = (long)(N_NODES / 16) * (OUT_DIM / 16);
        gemm_wmma_f32<<<cdiv(waves * 32, TB), TB, 0, stream>>>(
            agg1, W2, g2, N_NODES, HID, OUT_DIM);
    }
    alph_kernel<<<cdiv(N_NODES, TB), TB, 0, stream>>>(
        g2, a2s, a2d, as2, ad2, N_NODES, 1, OUT_DIM);
    edge_max<<<cdiv(E_TOT, TB), TB, 0, stream>>>(as2, ad2, src, dst, m2, 1);
    edge_exp<<<cdiv(E_TOT, TB), TB, 0, stream>>>(as2, ad2, src, dst, m2, s2, exb, 1);
    edge_agg<<<E_TOT, OUT_DIM, 0, stream>>>(g2, src, dst, exb, s2, out, 1, OUT_DIM);
    bias_add_inplace<<<cdiv((long)N_NODES * OUT_DIM, TB), TB, 0, stream>>>(
        out, b2, N_NODES * OUT_DIM, OUT_DIM);
}